// MatrixGraphConvolution_39032662786123
// MI455X (gfx1250) — compile-verified
//
#include <hip/hip_runtime.h>
#include <hip/hip_bf16.h>
#include <stdint.h>

#define F 128  // F_IN == F_OUT == 128

typedef float v2f __attribute__((ext_vector_type(2)));
typedef float v8f __attribute__((ext_vector_type(8)));

// ---------------------------------------------------------------------------
// Phase 0: zero a span of workspace (grid-stride, dword stores)
// ---------------------------------------------------------------------------
__global__ void gcn_zero_kernel(uint32_t* __restrict__ p, size_t nwords) {
    size_t i = (size_t)blockIdx.x * blockDim.x + threadIdx.x;
    size_t stride = (size_t)gridDim.x * blockDim.x;
    for (; i < nwords; i += stride) p[i] = 0u;
}

// ---------------------------------------------------------------------------
// Phase 1a: per-edge dedup mark. A[dst,src] has SET semantics, so only the
// first (dst,src) occurrence counts: atomicOr on an N*N bitmap decides the
// winner, which bumps deg[dst] and flags the edge as unique.
// ---------------------------------------------------------------------------
__global__ void gcn_edge_mark_kernel(const int* __restrict__ ei,
                                     float* __restrict__ deg,
                                     uint32_t* __restrict__ bitmap,
                                     uint8_t* __restrict__ uniq,
                                     int E, int N, int dedup) {
    int e = blockIdx.x * blockDim.x + threadIdx.x;
    if (e >= E) return;
    int s = ei[e];        // source (row 0 of edge_index)
    int d = ei[E + e];    // dest   (row 1 of edge_index)
    int is_unique = 1;
    if (dedup) {
        size_t bit = (size_t)d * (size_t)N + (size_t)s;
        uint32_t mask = 1u << (bit & 31u);
        uint32_t old = atomicOr(&bitmap[bit >> 5], mask);
        is_unique = (old & mask) ? 0 : 1;
    }
    uniq[e] = (uint8_t)is_unique;
    if (is_unique) atomicAdd(&deg[d], 1.0f);
}

// ---------------------------------------------------------------------------
// Phase 1b: one wave32 per edge; lanes cover the 128-float feature row in
// float4 chunks. agg[dst] += x[src] for unique edges only. x row reads are
// fully coalesced (512B per wave) and resident in L2 (x is only 8 MB).
// ---------------------------------------------------------------------------
__global__ void gcn_edge_scatter_kernel(const int* __restrict__ ei,
                                        const float* __restrict__ x,
                                        float* __restrict__ agg,
                                        const uint8_t* __restrict__ uniq,
                                        int E) {
    uint32_t tid = blockIdx.x * blockDim.x + threadIdx.x;
    int e = (int)(tid >> 5);           // wave-uniform: 32 consecutive lanes
    if (e >= E) return;
    if (!uniq[e]) return;
    int s = ei[e];
    int d = ei[E + e];
    int f = (int)(tid & 31u) << 2;     // 4 floats per lane
    const float4 v = *(const float4*)(x + (size_t)s * F + f);
    float* a = agg + (size_t)d * F + f;
    atomicAdd(a + 0, v.x);
    atomicAdd(a + 1, v.y);
    atomicAdd(a + 2, v.z);
    atomicAdd(a + 3, v.w);
}

// ---------------------------------------------------------------------------
// Phase 2: out = (agg * (1/deg)) @ W^T + x @ B^T via V_WMMA_F32_16X16X4_F32.
//
// CDNA5 data path: the block's shared 16-row A-stripes (agg rows, x rows) are
// staged into LDS ONCE via GLOBAL_LOAD_ASYNC_TO_LDS_B128 (ASYNCcnt-tracked,
// no VGPR round-trip), then all 8 waves read their A fragments from LDS
// (ds_load_b64) instead of redundantly re-loading them 8x from global.
// W/B fragments stay global: disjoint 16-row slices per wave, L2-resident.
//
// f32 WMMA fragment layout (wave32):
//   A (16x4): lane L -> row m0+(L&15); K pair {k,k+1} lanes 0-15,
//             {k+2,k+3} lanes 16-31  => contiguous float2 per lane.
//   B (4x16) = W^T tile: element (kk, o) = W[o0+o, kk]; mirrored lane layout
//             => contiguous float2 from row-major W as well.
//   C/D (16x16): VGPR r -> row m0 + r + (lane>=16 ? 8 : 0), col o0+(lane&15).
// ---------------------------------------------------------------------------
__global__ void gcn_wmma_gemm_kernel(const float* __restrict__ agg,
                                     const float* __restrict__ deg,
                                     const float* __restrict__ x,
                                     const float* __restrict__ W,
                                     const float* __restrict__ Bm,
                                     float* __restrict__ out) {
    __shared__ __align__(16) float lds_y[16 * F];   // 8 KB: agg stripe
    __shared__ __align__(16) float lds_x[16 * F];   // 8 KB: x   stripe

    const int t    = threadIdx.x;
    const int lane = t & 31;
    const int wave = t >> 5;                  // 0..7 -> output-col tile
    const int m0   = blockIdx.x * 16;
    const int o0   = wave * 16;
    const int row  = lane & 15;
    const int khi  = (lane >> 4) * 2;         // 0 for lanes 0-15, 2 for 16-31

    // ---- async-stage the A stripes: 2 x 8 KB = 2 x 512 16B chunks ---------
    {
        const char* yg = (const char*)(agg + (size_t)m0 * F);
        const char* xg = (const char*)(x   + (size_t)m0 * F);
        // flat pointers to LDS carry the logical LDS byte offset in the low
        // 32 bits (ISA 10.2 aperture mapping) -> usable as async VDST operand
        const uint32_t ylo = (uint32_t)(uintptr_t)lds_y;
        const uint32_t xlo = (uint32_t)(uintptr_t)lds_x;
#pragma unroll
        for (int c0 = 0; c0 < 512; c0 += 256) {
            const uint32_t c = (uint32_t)(c0 + t);
            asm volatile("global_load_async_to_lds_b128 %0, %1, off"
                         :: "v"(ylo + c * 16u), "v"(yg + (size_t)c * 16)
                         : "memory");
            asm volatile("global_load_async_to_lds_b128 %0, %1, off"
                         :: "v"(xlo + c * 16u), "v"(xg + (size_t)c * 16)
                         : "memory");
        }
        asm volatile("s_wait_asynccnt 0" ::: "memory");
    }
    __syncthreads();

    const float rd = 1.0f / deg[m0 + row];    // fold D^-1 into the A fragment

    const float* ylds = lds_y + row * F;
    const float* xlds = lds_x + row * F;
    const float* wrow = W  + (size_t)(o0 + row) * F;
    const float* brow = Bm + (size_t)(o0 + row) * F;

    v8f c = {0.f, 0.f, 0.f, 0.f, 0.f, 0.f, 0.f, 0.f};

    // (agg/deg) @ W^T : 32 WMMAs over K=128
#pragma unroll
    for (int k = 0; k < F; k += 4) {
        v2f a = *(const v2f*)(ylds + k + khi);      // ds_load_b64
        a.x *= rd;
        a.y *= rd;
        v2f b = *(const v2f*)(wrow + k + khi);      // global_load_b64
        c = __builtin_amdgcn_wmma_f32_16x16x4_f32(
                /*neg_a=*/false, a, /*neg_b=*/false, b,
                /*c_mod=*/(short)0, c, /*reuse_a=*/false, /*reuse_b=*/false);
    }
    // + x @ B^T : 32 more WMMAs into the same accumulator
#pragma unroll
    for (int k = 0; k < F; k += 4) {
        v2f a = *(const v2f*)(xlds + k + khi);
        v2f b = *(const v2f*)(brow + k + khi);
        c = __builtin_amdgcn_wmma_f32_16x16x4_f32(
                false, a, false, b, (short)0, c, false, false);
    }

    const int col   = o0 + (lane & 15);
    const int rbase = m0 + (lane >> 4) * 8;
#pragma unroll
    for (int r = 0; r < 8; ++r) {
        out[(size_t)(rbase + r) * F + col] = c[r];
    }
}

// ---------------------------------------------------------------------------
extern "C" void kernel_launch(void* const* d_in, const int* in_sizes, int n_in,
                              void* d_out, int out_size, void* d_ws, size_t ws_size,
                              hipStream_t stream) {
    const float* x  = (const float*)d_in[0];       // [N, 128]
    const int*   ei = (const int*)d_in[1];         // [2, E] row-major
    const float* W  = (const float*)d_in[2];       // [128, 128]
    const float* Bm = (const float*)d_in[3];       // [128, 128]
    float* out = (float*)d_out;                    // [N, 128]

    const int N = in_sizes[0] / F;
    const int E = in_sizes[1] / 2;

    // Workspace layout: [agg N*128 f32][deg N f32][uniq E u8][bitmap N*N bits]
    char* ws = (char*)d_ws;
    const size_t agg_bytes    = (size_t)N * F * sizeof(float);
    const size_t deg_off      = agg_bytes;
    const size_t deg_bytes    = (size_t)N * sizeof(float);
    const size_t uniq_off     = deg_off + deg_bytes;
    const size_t uniq_bytes   = (size_t)E;
    const size_t bitmap_off   = (uniq_off + uniq_bytes + 63) & ~(size_t)63;
    const size_t bitmap_bytes = (((size_t)N * (size_t)N + 7) / 8 + 3) & ~(size_t)3;
    const int dedup = (ws_size >= bitmap_off + bitmap_bytes) ? 1 : 0;

    float*    agg    = (float*)ws;
    float*    deg    = (float*)(ws + deg_off);
    uint8_t*  uniq   = (uint8_t*)(ws + uniq_off);
    uint32_t* bitmap = (uint32_t*)(ws + bitmap_off);

    // zero agg+deg (contiguous), and bitmap if used
    gcn_zero_kernel<<<2048, 256, 0, stream>>>((uint32_t*)ws,
                                              (agg_bytes + deg_bytes) / 4);
    if (dedup) {
        gcn_zero_kernel<<<8192, 256, 0, stream>>>(bitmap, bitmap_bytes / 4);
    }

    gcn_edge_mark_kernel<<<(E + 255) / 256, 256, 0, stream>>>(
        ei, deg, bitmap, uniq, E, N, dedup);

    gcn_edge_scatter_kernel<<<(E * 32 + 255) / 256, 256, 0, stream>>>(
        ei, x, agg, uniq, E);

    gcn_wmma_gemm_kernel<<<N / 16, 256, 0, stream>>>(agg, deg, x, W, Bm, out);
}